// SkinningModule_34059090657340
// MI455X (gfx1250) — compile-verified
//
#include <hip/hip_runtime.h>
#include <math.h>

#define L_DIM   4096
#define K_DIM   64
#define D_DIM   1024
#define HID_DIM 1024

typedef __attribute__((ext_vector_type(2))) float v2f;
typedef __attribute__((ext_vector_type(8))) float v8f;

// ---------------------------------------------------------------------------
// Kernel A: preTbT4[h4][k][j] = (T_all @ W1t)[k][4*h4+j] + b1[4*h4+j]
// Transposed+interleaved layout so kernel C gets coalesced float4 loads.
// grid = K * HID/256 = 256 blocks, 256 threads.
// ---------------------------------------------------------------------------
__global__ __launch_bounds__(256) void preT_kernel(
    const float* __restrict__ T_all,   // [K, D]
    const float* __restrict__ W1t,     // [D, HID]
    const float* __restrict__ b1,      // [HID]
    float* __restrict__ preTbT4)       // [HID/4][K][4]
{
  const int k = blockIdx.x >> 2;                       // 0..63
  const int h = ((blockIdx.x & 3) << 8) + threadIdx.x; // 0..1023
  const float* __restrict__ trow = T_all + (size_t)k * D_DIM;
  float acc = b1[h];
  #pragma unroll 4
  for (int d = 0; d < D_DIM; ++d) {
    acc = fmaf(trow[d], W1t[(size_t)d * HID_DIM + h], acc);
  }
  preTbT4[(((size_t)(h >> 2) * K_DIM + k) << 2) + (h & 3)] = acc;
}

// ---------------------------------------------------------------------------
// Kernel B: preH = H @ W1h via V_WMMA_F32_16X16X4_F32.
// One wave32 per 16x16 output tile; loop over D in steps of 4.
// A (16x4):  lane 0-15 -> row=lane, K={0,1}; lane 16-31 -> row=lane-16, K={2,3}
// B (4x16):  lane 0-15 -> col=lane, K={0,1}; lane 16-31 -> col=lane-16, K={2,3}
// C/D (16x16): lane 0-15 -> col=lane rows 0..7; lane 16-31 -> col rows 8..15
// grid = (L/16)*(HID/16) = 16384 blocks, 32 threads.
// ---------------------------------------------------------------------------
__global__ __launch_bounds__(32) void preH_wmma_kernel(
    const float* __restrict__ H,       // [L, D]
    const float* __restrict__ W1h,     // [D, HID]
    float* __restrict__ preH)          // [L, HID]
{
  const int tiles_n = HID_DIM / 16;            // 64
  const int tm = blockIdx.x / tiles_n;         // 0..255
  const int tn = blockIdx.x % tiles_n;         // 0..63
  const int lane = threadIdx.x & 31;
  const int half = lane >> 4;                  // 0 or 1
  const int l15  = lane & 15;

  const int arow = tm * 16 + l15;              // A row held by this lane
  const int bcol = tn * 16 + l15;              // B col held by this lane

  const float* __restrict__ Arow = H + (size_t)arow * D_DIM + 2 * half;

  v8f c = {0.f, 0.f, 0.f, 0.f, 0.f, 0.f, 0.f, 0.f};

  for (int kk = 0; kk < D_DIM; kk += 4) {
    // A operand: two consecutive f32 (8B aligned) -> single b64 load
    v2f a = *reinterpret_cast<const v2f*>(Arow + kk);
    // B operand: rows kk+2*half, kk+2*half+1 of W1h at column bcol
    v2f b;
    b.x = W1h[(size_t)(kk + 2 * half + 0) * HID_DIM + bcol];
    b.y = W1h[(size_t)(kk + 2 * half + 1) * HID_DIM + bcol];
    c = __builtin_amdgcn_wmma_f32_16x16x4_f32(
        /*neg_a=*/false, a, /*neg_b=*/false, b,
        /*c_mod=*/(short)0, c, /*reuse_a=*/false, /*reuse_b=*/false);
  }

  float* __restrict__ out = preH + (size_t)(tm * 16 + half * 8) * HID_DIM + tn * 16 + l15;
  #pragma unroll
  for (int r = 0; r < 8; ++r) {
    out[(size_t)r * HID_DIM] = c[r];
  }
}

// ---------------------------------------------------------------------------
// Kernel C: per-row fused relu-dot + softmax.
// One block per l (4096 blocks), 64 threads; thread t == k.
// preH row + w2 staged in LDS (broadcast reads); preT table read as
// coalesced float4 (lane t reads [h4][t][0..3]).
// ---------------------------------------------------------------------------
__global__ __launch_bounds__(64) void score_softmax_kernel(
    const float* __restrict__ preH,     // [L, HID]
    const float* __restrict__ preTbT4,  // [HID/4][K][4]
    const float* __restrict__ w2,       // [HID]
    const float* __restrict__ b2,       // [1]
    float* __restrict__ out)            // [L, K]
{
  __shared__ float4 sh_preH[HID_DIM / 4];
  __shared__ float4 sh_w2[HID_DIM / 4];
  __shared__ float  sh_logit[K_DIM];
  __shared__ float  sh_exp[K_DIM];

  const int l = blockIdx.x;
  const int t = threadIdx.x;   // == k

  const float4* __restrict__ pH4 = reinterpret_cast<const float4*>(preH + (size_t)l * HID_DIM);
  const float4* __restrict__ pw4 = reinterpret_cast<const float4*>(w2);
  #pragma unroll
  for (int i = t; i < HID_DIM / 4; i += K_DIM) {
    sh_preH[i] = pH4[i];
    sh_w2[i]   = pw4[i];
  }
  __syncthreads();

  const float4* __restrict__ pT4 = reinterpret_cast<const float4*>(preTbT4);
  float acc = 0.f;
  #pragma unroll 4
  for (int h4 = 0; h4 < HID_DIM / 4; ++h4) {
    float4 tv = pT4[(size_t)h4 * K_DIM + t];   // coalesced 16B per lane
    float4 ph = sh_preH[h4];                   // broadcast
    float4 wv = sh_w2[h4];                     // broadcast
    acc = fmaf(fmaxf(ph.x + tv.x, 0.f), wv.x, acc);
    acc = fmaf(fmaxf(ph.y + tv.y, 0.f), wv.y, acc);
    acc = fmaf(fmaxf(ph.z + tv.z, 0.f), wv.z, acc);
    acc = fmaf(fmaxf(ph.w + tv.w, 0.f), wv.w, acc);
  }

  const float logit = acc + b2[0];
  sh_logit[t] = logit;
  __syncthreads();

  float m = sh_logit[0];
  #pragma unroll
  for (int j = 1; j < K_DIM; ++j) m = fmaxf(m, sh_logit[j]);

  const float e = __expf(logit - m);
  sh_exp[t] = e;
  __syncthreads();

  float s = 0.f;
  #pragma unroll
  for (int j = 0; j < K_DIM; ++j) s += sh_exp[j];

  out[(size_t)l * K_DIM + t] = e / s;
}

// ---------------------------------------------------------------------------
// Launch
// ---------------------------------------------------------------------------
extern "C" void kernel_launch(void* const* d_in, const int* in_sizes, int n_in,
                              void* d_out, int out_size, void* d_ws, size_t ws_size,
                              hipStream_t stream) {
  const float* H     = (const float*)d_in[0];  // [L, D]
  const float* T_all = (const float*)d_in[1];  // [K, D]
  const float* W1h   = (const float*)d_in[2];  // [D, HID]
  const float* W1t   = (const float*)d_in[3];  // [D, HID]
  const float* b1    = (const float*)d_in[4];  // [HID]
  const float* w2    = (const float*)d_in[5];  // [HID]
  const float* b2    = (const float*)d_in[6];  // [1]
  float* out = (float*)d_out;                  // [L, K]

  float* preTbT4 = (float*)d_ws;                       // 64*1024 f32 = 256 KB
  float* preH    = (float*)d_ws + (size_t)K_DIM * HID_DIM; // 4096*1024 f32 = 16 MB

  // Independent producers
  preT_kernel<<<K_DIM * (HID_DIM / 256), 256, 0, stream>>>(T_all, W1t, b1, preTbT4);
  preH_wmma_kernel<<<(L_DIM / 16) * (HID_DIM / 16), 32, 0, stream>>>(H, W1h, preH);
  // Consumer (stream-ordered)
  score_softmax_kernel<<<L_DIM, 64, 0, stream>>>(preH, preTbT4, w2, b2, out);
}